// DeformConv1d_86629490360665
// MI455X (gfx1250) — compile-verified
//
#include <hip/hip_runtime.h>
#include <math.h>

typedef float v2f __attribute__((ext_vector_type(2)));
typedef float v8f __attribute__((ext_vector_type(8)));

#define C_DIM 1024
#define L_DIM 2048
#define N_BATCH 4
#define NL (N_BATCH * L_DIM)
#define K_TAP 7

struct Samp { int pf; int pc; float wf; float wc; };

// ---------------------------------------------------------------------------
// GEMM: D[M,Nn] = A[M,K] @ W[Nn,K]^T + bias[Nn], exact fp32 via
// V_WMMA_F32_16X16X4_F32. One wave computes a 64x64 output tile.
// A fragment layout (16x4): lane = m + 16*(k>>1), vgpr = k&1
// B fragment layout (4x16): lane = n + 16*(k>>1), vgpr = k&1  (B = W^T, so
//   both fragments load 2 K-consecutive floats -> one b64 load each)
// D layout: row = v + 8*(lane>>4), col = lane&15
// ---------------------------------------------------------------------------
__global__ void __launch_bounds__(128)
gemm_f32_wmma_64x64(const float* __restrict__ A, const float* __restrict__ W,
                    const float* __restrict__ bias, float* __restrict__ D,
                    int M, int Nn, int Kk) {
  const int lane = threadIdx.x & 31;
  const int gw = blockIdx.x * 4 + (threadIdx.x >> 5);  // global wave id
  const int tilesN = Nn >> 6;
  const int Mbase = (gw / tilesN) * 64;
  const int Nbase = (gw % tilesN) * 64;
  const int mi = lane & 15;
  const int kh = lane >> 4;  // 0 or 1 -> K offset 2*kh

  v8f acc[4][4];
  const v8f vzero = {0.f, 0.f, 0.f, 0.f, 0.f, 0.f, 0.f, 0.f};
#pragma unroll
  for (int i = 0; i < 4; ++i)
#pragma unroll
    for (int j = 0; j < 4; ++j) acc[i][j] = vzero;

  const float* aP[4];
  const float* bP[4];
#pragma unroll
  for (int t = 0; t < 4; ++t) {
    aP[t] = A + (size_t)(Mbase + t * 16 + mi) * Kk + 2 * kh;
    bP[t] = W + (size_t)(Nbase + t * 16 + mi) * Kk + 2 * kh;
  }

  for (int kb = 0; kb < Kk; kb += 4) {
    v2f a[4], b[4];
#pragma unroll
    for (int t = 0; t < 4; ++t) {
      a[t] = *(const v2f*)(aP[t] + kb);
      b[t] = *(const v2f*)(bP[t] + kb);
    }
#pragma unroll
    for (int i = 0; i < 4; ++i)
#pragma unroll
      for (int j = 0; j < 4; ++j)
        acc[i][j] = __builtin_amdgcn_wmma_f32_16x16x4_f32(
            false, a[i], false, b[j], (short)0, acc[i][j], false, false);
  }

#pragma unroll
  for (int i = 0; i < 4; ++i) {
#pragma unroll
    for (int j = 0; j < 4; ++j) {
      const int col = Nbase + j * 16 + mi;
      const float bv = bias[col];
#pragma unroll
      for (int v = 0; v < 8; ++v) {
        const int row = Mbase + i * 16 + v + 8 * kh;
        D[(size_t)row * Nn + col] = acc[i][j][v] + bv;
      }
    }
  }
}

// ---------------------------------------------------------------------------
// Per-(n,l): depthwise conv(k=3,pad=1) on x -> LayerNorm -> exact GELU ->
// offset/mask heads (7 each) -> softmax -> sampling records.
// 256 threads, 4 channels per thread.
// ---------------------------------------------------------------------------
__global__ void __launch_bounds__(256)
dw_ln_heads_kernel(const float* __restrict__ x, const float* __restrict__ dw_w,
                   const float* __restrict__ dw_b, const float* __restrict__ ln_g,
                   const float* __restrict__ ln_b, const float* __restrict__ Wo,
                   const float* __restrict__ bo, const float* __restrict__ Wm,
                   const float* __restrict__ bm, Samp* __restrict__ samp) {
  __shared__ float red1[256];
  __shared__ float red2[256];
  __shared__ float part[256 * 16];  // 14 head partials per thread (stride 16)
  __shared__ float headv[16];
  __shared__ float stats[2];

  const int nl = blockIdx.x;
  const int l = nl % L_DIM;
  const int tid = threadIdx.x;
  const int c0 = tid * 4;
  const float* xr = x + (size_t)nl * C_DIM;

  float h[4];
  float s1 = 0.f, s2 = 0.f;
#pragma unroll
  for (int j = 0; j < 4; ++j) {
    const int c = c0 + j;
    const float xm = (l > 0) ? xr[c - C_DIM] : 0.0f;
    const float x0 = xr[c];
    const float xp = (l < L_DIM - 1) ? xr[c + C_DIM] : 0.0f;
    const float y = xm * dw_w[c * 3 + 0] + x0 * dw_w[c * 3 + 1] +
                    xp * dw_w[c * 3 + 2] + dw_b[c];
    h[j] = y;
    s1 += y;
    s2 += y * y;
  }
  red1[tid] = s1;
  red2[tid] = s2;
  __syncthreads();
  for (int off = 128; off > 0; off >>= 1) {
    if (tid < off) {
      red1[tid] += red1[tid + off];
      red2[tid] += red2[tid + off];
    }
    __syncthreads();
  }
  if (tid == 0) {
    const float mu = red1[0] * (1.0f / C_DIM);
    const float var = red2[0] * (1.0f / C_DIM) - mu * mu;
    stats[0] = mu;
    stats[1] = rsqrtf(var + 1e-5f);
  }
  __syncthreads();
  const float mu = stats[0], rs = stats[1];

  float acc14[14];
#pragma unroll
  for (int k = 0; k < 14; ++k) acc14[k] = 0.f;
#pragma unroll
  for (int j = 0; j < 4; ++j) {
    const int c = c0 + j;
    float v = (h[j] - mu) * rs * ln_g[c] + ln_b[c];
    v = 0.5f * v * (1.0f + erff(v * 0.70710678118654752f));  // exact GELU
#pragma unroll
    for (int k = 0; k < K_TAP; ++k) {
      acc14[k] += v * Wo[k * C_DIM + c];
      acc14[7 + k] += v * Wm[k * C_DIM + c];
    }
  }
#pragma unroll
  for (int k = 0; k < 14; ++k) part[tid * 16 + k] = acc14[k];
  __syncthreads();
  if (tid < 14) {
    float s = 0.f;
    for (int i = 0; i < 256; ++i) s += part[i * 16 + tid];
    headv[tid] = s;
  }
  __syncthreads();
  if (tid == 0) {
    float off7[K_TAP], mg[K_TAP];
    float mmax = -1e30f;
    for (int k = 0; k < K_TAP; ++k) {
      off7[k] = (headv[k] + bo[k]) * 2.0f;  // OFFSET_SCALE
      mg[k] = headv[7 + k] + bm[k];
      mmax = fmaxf(mmax, mg[k]);
    }
    float msum = 0.f;
    for (int k = 0; k < K_TAP; ++k) {
      mg[k] = expf(mg[k] - mmax);
      msum += mg[k];
    }
    const float inv = 1.0f / msum;
    for (int k = 0; k < K_TAP; ++k) {
      float m = mg[k] * inv;
      if (!(m == m)) m = 0.0f;  // nan_to_num(nan=0)
      const float ap = (float)l + (float)(k - 3) + off7[k];
      float apc = fminf(fmaxf(ap, 0.0f), (float)(L_DIM - 1));
      int pf = (int)apc;
      if (pf < 0) pf = 0;
      if (pf > L_DIM - 1) pf = L_DIM - 1;
      int pc = pf + 1;
      if (pc > L_DIM - 1) pc = L_DIM - 1;
      float wc = apc - (float)pf;
      float wf = 1.0f - wc;
      const float valid = (ap < 0.0f || ap > (float)(L_DIM - 1)) ? 0.0f : 1.0f;
      wf *= valid * m;
      wc *= valid * m;
      Samp sp;
      sp.pf = pf;
      sp.pc = pc;
      sp.wf = wf;
      sp.wc = wc;
      samp[nl * K_TAP + k] = sp;
    }
  }
}

// ---------------------------------------------------------------------------
// Deformable gather: mid[n,l,:] = sum_k wf*x_proj[n,pf,:] + wc*x_proj[n,pc,:]
// x_proj is L2-resident (32MB << 192MB); float4 per thread, coalesced.
// ---------------------------------------------------------------------------
__global__ void __launch_bounds__(256)
gather_kernel(const float* __restrict__ xproj, const Samp* __restrict__ samp,
              float* __restrict__ mid) {
  __shared__ Samp sp[K_TAP];
  const int nl = blockIdx.x;
  const int n = nl / L_DIM;
  const int tid = threadIdx.x;
  if (tid < K_TAP) sp[tid] = samp[nl * K_TAP + tid];
  __syncthreads();
  const int c0 = tid * 4;
  float4 acc = {0.f, 0.f, 0.f, 0.f};
#pragma unroll
  for (int k = 0; k < K_TAP; ++k) {
    const float4 vf =
        *(const float4*)(xproj + ((size_t)(n * L_DIM + sp[k].pf)) * C_DIM + c0);
    const float4 vc =
        *(const float4*)(xproj + ((size_t)(n * L_DIM + sp[k].pc)) * C_DIM + c0);
    const float wf = sp[k].wf, wc = sp[k].wc;
    acc.x += wf * vf.x + wc * vc.x;
    acc.y += wf * vf.y + wc * vc.y;
    acc.z += wf * vf.z + wc * vc.z;
    acc.w += wf * vf.w + wc * vc.w;
  }
  *(float4*)(mid + (size_t)nl * C_DIM + c0) = acc;
}

// ---------------------------------------------------------------------------
extern "C" void kernel_launch(void* const* d_in, const int* in_sizes, int n_in,
                              void* d_out, int out_size, void* d_ws,
                              size_t ws_size, hipStream_t stream) {
  const float* x = (const float*)d_in[0];
  const float* Wi = (const float*)d_in[1];
  const float* bi = (const float*)d_in[2];
  const float* dw_w = (const float*)d_in[3];
  const float* dw_b = (const float*)d_in[4];
  const float* ln_g = (const float*)d_in[5];
  const float* ln_b = (const float*)d_in[6];
  const float* Wo = (const float*)d_in[7];
  const float* bo = (const float*)d_in[8];
  const float* Wm = (const float*)d_in[9];
  const float* bm = (const float*)d_in[10];
  const float* Wout = (const float*)d_in[11];
  const float* bout = (const float*)d_in[12];
  float* out = (float*)d_out;

  char* ws = (char*)d_ws;
  float* xproj = (float*)ws;                              // 32 MB
  float* mid = (float*)(ws + (size_t)NL * C_DIM * 4);     // 32 MB
  Samp* samp = (Samp*)(ws + (size_t)NL * C_DIM * 8);      // ~0.9 MB

  const int waveTiles = (NL / 64) * (C_DIM / 64);  // 2048
  const int gemmBlocks = waveTiles / 4;            // 4 waves per block

  gemm_f32_wmma_64x64<<<gemmBlocks, 128, 0, stream>>>(x, Wi, bi, xproj, NL,
                                                      C_DIM, C_DIM);
  dw_ln_heads_kernel<<<NL, 256, 0, stream>>>(x, dw_w, dw_b, ln_g, ln_b, Wo, bo,
                                             Wm, bm, samp);
  gather_kernel<<<NL, 256, 0, stream>>>(xproj, samp, mid);
  gemm_f32_wmma_64x64<<<gemmBlocks, 128, 0, stream>>>(mid, Wout, bout, out, NL,
                                                      C_DIM, C_DIM);
}